// ngcf1_session_hot_items_88957362635442
// MI455X (gfx1250) — compile-verified
//
#include <hip/hip_runtime.h>

typedef __attribute__((ext_vector_type(2))) float v2f;
typedef __attribute__((ext_vector_type(8))) float v8f;

// Problem constants (fixed by the reference harness)
constexpr int EMB      = 128;      // EMB_DIM == HID_DIM
constexpr int N_ITEMS  = 50000;
constexpr int N_SESS   = 50000;
constexpr int N_TOTAL  = 100000;
constexpr int NNZ_S    = 500000;
constexpr int NNZ_A    = 3200000;
constexpr int BATCH    = 2048;
constexpr int N_SCORE  = 50000;

#define WMMA_F32(A, B, C) \
  __builtin_amdgcn_wmma_f32_16x16x4_f32(false, (A), false, (B), (short)0, (C), false, false)

// ---------------------------------------------------------------------------
// 1) x = [zeros(N_SESS,128) ; emb_table[item_emb_idxes]]   (float4 per thread)
// ---------------------------------------------------------------------------
__global__ __launch_bounds__(256) void k_build_x(
    const float* __restrict__ emb, const int* __restrict__ idx,
    float* __restrict__ x) {
  long tid = (long)blockIdx.x * blockDim.x + threadIdx.x;
  const long total = (long)N_TOTAL * (EMB / 4);
  if (tid >= total) return;
  int row = (int)(tid >> 5);       // EMB/4 == 32 float4 per row
  int c4  = (int)(tid & 31);
  float4 v;
  if (row < N_SESS) {
    v = make_float4(0.f, 0.f, 0.f, 0.f);
  } else {
    int src = idx[row - N_SESS];
    v = ((const float4*)(emb + (long)src * EMB))[c4];
  }
  ((float4*)(x + (long)row * EMB))[c4] = v;
}

// ---------------------------------------------------------------------------
// 2/5) SpMM via per-edge wave + f32 atomic scatter-add.
//      dst[rows[e]] += vals[e] * src[(cols[e]+col_off)]
// ---------------------------------------------------------------------------
__global__ __launch_bounds__(256) void k_spmm_atomic(
    const float* __restrict__ vals, const int* __restrict__ rows,
    const int* __restrict__ cols, const float* __restrict__ src,
    float* __restrict__ dst, int nnz, int col_off) {
  long tid = (long)blockIdx.x * blockDim.x + threadIdx.x;
  long e = tid >> 5;               // one wave32 per edge
  if (e >= nnz) return;
  int lane = (int)(tid & 31);      // 4 dims per lane
  int r = rows[e];
  int c = cols[e];
  float v = vals[e];
  float4 s = ((const float4*)(src + (long)(c + col_off) * EMB))[lane];
  float* d = dst + (long)r * EMB + lane * 4;
  atomicAdd(d + 0, v * s.x);
  atomicAdd(d + 1, v * s.y);
  atomicAdd(d + 2, v * s.z);
  atomicAdd(d + 3, v * s.w);
}

// ---------------------------------------------------------------------------
// 3) xw = x @ W : one wave per 32x16 tile (2 M-tiles share the W fragment)
// ---------------------------------------------------------------------------
__global__ __launch_bounds__(256) void k_gemm_xw(
    const float* __restrict__ x, const float* __restrict__ W,
    float* __restrict__ xw) {
  int wid  = (int)(((long)blockIdx.x * blockDim.x + threadIdx.x) >> 5);
  int lane = threadIdx.x & 31;
  constexpr int TN = EMB / 16;     // 8 column tiles
  int tm = wid / TN;               // covers rows [tm*32, tm*32+32)
  int tn = wid % TN;
  if (tm >= N_TOTAL / 32) return;  // wave-uniform

  int r    = lane & 15;
  int half = lane >> 4;            // 0: K={0,1}, 1: K={2,3}

  const float* a0p   = x + (long)(tm * 32 + r) * EMB + half * 2;
  const float* a1p   = x + (long)(tm * 32 + 16 + r) * EMB + half * 2;
  const float* bbase = W + (long)(half * 2) * EMB + tn * 16 + r;

  v8f c0 = {}, c1 = {};
#pragma unroll 4
  for (int k = 0; k < EMB; k += 4) {
    v2f a0 = *(const v2f*)(a0p + k);
    v2f a1 = *(const v2f*)(a1p + k);
    v2f b;
    b.x = bbase[(long)k * EMB];          // W[k+2*half  ][n]
    b.y = bbase[(long)k * EMB + EMB];    // W[k+2*half+1][n]
    c0 = WMMA_F32(a0, b, c0);
    c1 = WMMA_F32(a1, b, c1);
  }

  float* o0 = xw + (long)(tm * 32 + half * 8) * EMB + tn * 16 + r;
  float* o1 = xw + (long)(tm * 32 + 16 + half * 8) * EMB + tn * 16 + r;
#pragma unroll
  for (int v = 0; v < 8; ++v) {
    o0[(long)v * EMB] = c0[v];
    o1[(long)v * EMB] = c1[v];
  }
}

// ---------------------------------------------------------------------------
// 4) h1 = broadcast(bias)   (h1 aliases x; x is dead after k_gemm_xw)
// ---------------------------------------------------------------------------
__global__ __launch_bounds__(256) void k_init_h1(
    const float* __restrict__ b, float* __restrict__ h1) {
  long tid = (long)blockIdx.x * blockDim.x + threadIdx.x;
  const long total = (long)N_TOTAL * (EMB / 4);
  if (tid >= total) return;
  int c4 = (int)(tid & 31);
  float4 bv = ((const float4*)b)[c4];
  ((float4*)h1)[tid] = bv;
}

// ---------------------------------------------------------------------------
// 6) out[2048,50000] = h1[batch] @ h1[items]^T
//    2x2 register-blocked: one wave per 32x32 tile (4 WMMAs per 4 loads).
//    N has 3125 (odd) 16-tiles: last pair clamps its gather (loads stay
//    in-bounds, EXEC stays all-ones) and skips the invalid stores.
// ---------------------------------------------------------------------------
__global__ __launch_bounds__(256) void k_gemm_score(
    const float* __restrict__ h1, const int* __restrict__ bidx,
    const int* __restrict__ iidx, float* __restrict__ out) {
  int wid  = (int)(((long)blockIdx.x * blockDim.x + threadIdx.x) >> 5);
  int lane = threadIdx.x & 31;
  constexpr int NT  = N_SCORE / 16;      // 3125
  constexpr int NT2 = (NT + 1) / 2;      // 1563 tile pairs
  int tm = wid / NT2;                    // covers rows [tm*32, tm*32+32)
  int tp = wid % NT2;                    // covers cols [tp*32, tp*32+32)
  if (tm >= BATCH / 32) return;          // wave-uniform

  int r    = lane & 15;
  int half = lane >> 4;

  int m0 = tm * 32;
  int n0 = tp * 32;
  bool valid1 = (n0 + 16) < N_SCORE;     // wave-uniform

  int qr0 = bidx[m0 + r];
  int qr1 = bidx[m0 + 16 + r];
  int pr0 = iidx[n0 + r];
  int pr1 = iidx[valid1 ? (n0 + 16 + r) : (n0 + r)];  // clamp keeps loads legal

  const float* a0p = h1 + (long)qr0 * EMB + half * 2;
  const float* a1p = h1 + (long)qr1 * EMB + half * 2;
  const float* b0p = h1 + (long)pr0 * EMB + half * 2;
  const float* b1p = h1 + (long)pr1 * EMB + half * 2;

  v8f c00 = {}, c01 = {}, c10 = {}, c11 = {};
#pragma unroll 4
  for (int k = 0; k < EMB; k += 4) {
    v2f a0 = *(const v2f*)(a0p + k);
    v2f a1 = *(const v2f*)(a1p + k);
    v2f b0 = *(const v2f*)(b0p + k);
    v2f b1 = *(const v2f*)(b1p + k);
    c00 = WMMA_F32(a0, b0, c00);
    c01 = WMMA_F32(a0, b1, c01);
    c10 = WMMA_F32(a1, b0, c10);
    c11 = WMMA_F32(a1, b1, c11);
  }

  float* o00 = out + (long)(m0 + half * 8) * N_SCORE + n0 + r;
  float* o10 = out + (long)(m0 + 16 + half * 8) * N_SCORE + n0 + r;
#pragma unroll
  for (int v = 0; v < 8; ++v) {
    o00[(long)v * N_SCORE] = c00[v];
    o10[(long)v * N_SCORE] = c10[v];
  }
  if (valid1) {
#pragma unroll
    for (int v = 0; v < 8; ++v) {
      o00[(long)v * N_SCORE + 16] = c01[v];
      o10[(long)v * N_SCORE + 16] = c11[v];
    }
  }
}

// ---------------------------------------------------------------------------
extern "C" void kernel_launch(void* const* d_in, const int* in_sizes, int n_in,
                              void* d_out, int out_size, void* d_ws, size_t ws_size,
                              hipStream_t stream) {
  const float* emb_table      = (const float*)d_in[0];
  const float* W              = (const float*)d_in[1];
  const float* b              = (const float*)d_in[2];
  const float* sess_vals      = (const float*)d_in[3];
  const float* A_vals         = (const float*)d_in[4];
  const int*   batch_idxes    = (const int*)d_in[5];
  const int*   item_idxes     = (const int*)d_in[6];
  const int*   item_emb_idxes = (const int*)d_in[7];
  const int*   sess_rows      = (const int*)d_in[8];
  const int*   sess_cols      = (const int*)d_in[9];
  const int*   A_rows         = (const int*)d_in[10];
  const int*   A_cols         = (const int*)d_in[11];
  float*       out            = (float*)d_out;

  // Workspace: x (reused as h1) | xw   -> 2 * 100000*128*4 B = 102.4 MB
  float* x  = (float*)d_ws;
  float* xw = x + (long)N_TOTAL * EMB;
  float* h1 = x;                   // alias: x dead after k_gemm_xw

  // 1) x = [0 ; gather(emb_table)]
  {
    long thr = (long)N_TOTAL * (EMB / 4);
    k_build_x<<<(int)((thr + 255) / 256), 256, 0, stream>>>(emb_table, item_emb_idxes, x);
  }
  // 2) x[:N_SESS] += spmm(sess, x_item)
  {
    long thr = (long)NNZ_S * 32;
    k_spmm_atomic<<<(int)((thr + 255) / 256), 256, 0, stream>>>(
        sess_vals, sess_rows, sess_cols, x, x, NNZ_S, N_SESS);
  }
  // 3) xw = x @ W   (WMMA f32, 32x16 per wave)
  {
    long waves = (long)(N_TOTAL / 32) * (EMB / 16);   // 25000
    k_gemm_xw<<<(int)((waves * 32 + 255) / 256), 256, 0, stream>>>(x, W, xw);
  }
  // 4) h1 = bias (overwrites x)
  {
    long thr = (long)N_TOTAL * (EMB / 4);
    k_init_h1<<<(int)((thr + 255) / 256), 256, 0, stream>>>(b, h1);
  }
  // 5) h1 += spmm(A, xw)
  {
    long thr = (long)NNZ_A * 32;
    k_spmm_atomic<<<(int)((thr + 255) / 256), 256, 0, stream>>>(
        A_vals, A_rows, A_cols, xw, h1, NNZ_A, 0);
  }
  // 6) out = h1[batch] @ h1[items]^T   (WMMA f32, 32x32 per wave)
  {
    constexpr long NT2 = (N_SCORE / 16 + 1) / 2;      // 1563
    long waves = (long)(BATCH / 32) * NT2;            // 100032
    k_gemm_score<<<(int)((waves * 32 + 255) / 256), 256, 0, stream>>>(
        h1, batch_idxes, item_idxes, out);
  }
}